// GCN_463856468564
// MI455X (gfx1250) — compile-verified
//
#include <hip/hip_runtime.h>
#include <hip/hip_bf16.h>

typedef float v2f __attribute__((ext_vector_type(2)));
typedef float v8f __attribute__((ext_vector_type(8)));

static constexpr int NN    = 250000;
static constexpr int NE    = 4000000;
static constexpr int NTILE = NN / 16;   // 15625, exact

// ---------- degree / norm ----------
__global__ void k_deg_init(float* __restrict__ deg) {
    int i = blockIdx.x * blockDim.x + threadIdx.x;
    if (i < NN) deg[i] = 1.0f;   // self-loop contributes 1 to in-degree
}

__global__ void k_deg_count(const int* __restrict__ col, float* __restrict__ deg) {
    int i = blockIdx.x * blockDim.x + threadIdx.x;
    if (i < NE) atomicAdd(&deg[col[i]], 1.0f);
}

__global__ void k_dis(float* __restrict__ deg) {
    int i = blockIdx.x * blockDim.x + threadIdx.x;
    if (i < NN) {
        float d = deg[i];
        deg[i] = (d > 0.0f) ? rsqrtf(d) : 0.0f;   // in-place: deg -> dis
    }
}

// ---------- layer 1 linear: h1[N,16] = x[N,3] @ W1[3,16], f32 WMMA K=4 (K padded) ----------
__global__ void k_lin1_wmma(const float* __restrict__ x, const float* __restrict__ W1,
                            float* __restrict__ h1) {
    int wave = (blockIdx.x * blockDim.x + threadIdx.x) >> 5;   // one 16-node tile per wave32
    if (wave >= NTILE) return;                                  // wave-uniform exit
    int lane = threadIdx.x & 31;
    int half = lane >> 4;     // 0: K={0,1}, 1: K={2,3}
    int nidx = lane & 15;     // A row (M) and B column (N)
    int node = wave * 16 + nidx;

    v2f a, b;
    // A 16x4 layout: lane holds M=nidx; vgpr0 K=2*half, vgpr1 K=2*half+1 (K=3 padded with 0)
    a.x = half ? x[node * 3 + 2] : x[node * 3 + 0];
    a.y = half ? 0.0f            : x[node * 3 + 1];
    // B 4x16 layout: lane holds N=nidx; vgpr0 K=2*half, vgpr1 K=2*half+1
    b.x = half ? W1[2 * 16 + nidx] : W1[0 * 16 + nidx];
    b.y = half ? 0.0f              : W1[1 * 16 + nidx];

    v8f c = {};
    c = __builtin_amdgcn_wmma_f32_16x16x4_f32(false, a, false, b, (short)0, c, false, false);

#pragma unroll
    for (int r = 0; r < 8; ++r) {
        int m = r + 8 * half;                       // D row
        h1[(wave * 16 + m) * 16 + nidx] = c[r];     // D col = nidx
    }
}

// ---------- self-loop init: agg1[i,f] = dis[i]^2 * h1[i,f] ----------
__global__ void k_agg1_init(const float* __restrict__ h1, const float* __restrict__ dis,
                            float* __restrict__ agg) {
    int i = blockIdx.x * blockDim.x + threadIdx.x;
    if (i < NN * 16) {
        int node = i >> 4;
        float d = dis[node];
        agg[i] = d * d * h1[i];
    }
}

// ---------- edge scatter, 16 features, 4 threads/edge ----------
__global__ void k_edge1(const int* __restrict__ row, const int* __restrict__ col,
                        const float* __restrict__ dis, const float* __restrict__ h1,
                        float* __restrict__ agg) {
    long long t = (long long)blockIdx.x * blockDim.x + threadIdx.x;
    if (t >= (long long)NE * 4) return;
    int e = (int)(t >> 2), q = (int)(t & 3);
    int r = row[e], c = col[e];
    float w = dis[r] * dis[c];
    const float4 v = *(const float4*)(h1 + (long long)r * 16 + q * 4);
    float* dst = agg + (long long)c * 16 + q * 4;
    atomicAdd(dst + 0, w * v.x);
    atomicAdd(dst + 1, w * v.y);
    atomicAdd(dst + 2, w * v.z);
    atomicAdd(dst + 3, w * v.w);
}

// ---------- bias + relu: hrelu[i,f] = max(agg1[i,f] + b1[f], 0) ----------
__global__ void k_relu1(const float* __restrict__ agg, const float* __restrict__ b1,
                        float* __restrict__ hrelu) {
    int i = blockIdx.x * blockDim.x + threadIdx.x;
    if (i < NN * 16) {
        int f = i & 15;
        hrelu[i] = fmaxf(agg[i] + b1[f], 0.0f);
    }
}

// ---------- layer 2 linear: h2[N,16(pad)] = hrelu[N,16] @ W2pad[16,16], 4 chained f32 WMMA ----------
__global__ void k_lin2_wmma(const float* __restrict__ h, const float* __restrict__ W2,
                            float* __restrict__ h2) {
    int wave = (blockIdx.x * blockDim.x + threadIdx.x) >> 5;
    if (wave >= NTILE) return;
    int lane = threadIdx.x & 31;
    int half = lane >> 4;
    int nidx = lane & 15;
    int node = wave * 16 + nidx;

    v8f c = {};
#pragma unroll
    for (int ko = 0; ko < 16; ko += 4) {
        int k0 = ko + 2 * half;
        v2f a, b;
        a.x = h[node * 16 + k0];
        a.y = h[node * 16 + k0 + 1];
        b.x = (nidx < 7) ? W2[k0 * 7 + nidx] : 0.0f;        // pad cols 7..15 with 0
        b.y = (nidx < 7) ? W2[(k0 + 1) * 7 + nidx] : 0.0f;
        c = __builtin_amdgcn_wmma_f32_16x16x4_f32(false, a, false, b, (short)0, c, false, false);
    }
#pragma unroll
    for (int r = 0; r < 8; ++r) {
        int m = r + 8 * half;
        h2[(wave * 16 + m) * 16 + nidx] = c[r];
    }
}

// ---------- out init: out[i,f] = b2[f] + dis[i]^2 * h2[i,f]  (fully overwrites d_out) ----------
__global__ void k_out_init(const float* __restrict__ h2, const float* __restrict__ dis,
                           const float* __restrict__ b2, float* __restrict__ out) {
    int i = blockIdx.x * blockDim.x + threadIdx.x;
    if (i < NN * 7) {
        int node = i / 7;
        int f = i - node * 7;
        float d = dis[node];
        out[i] = b2[f] + d * d * h2[node * 16 + f];
    }
}

// ---------- edge scatter, 7 features, 2 threads/edge ----------
__global__ void k_edge2(const int* __restrict__ row, const int* __restrict__ col,
                        const float* __restrict__ dis, const float* __restrict__ h2,
                        float* __restrict__ out) {
    long long t = (long long)blockIdx.x * blockDim.x + threadIdx.x;
    if (t >= (long long)NE * 2) return;
    int e = (int)(t >> 1), q = (int)(t & 1);
    int r = row[e], c = col[e];
    float w = dis[r] * dis[c];
    const float4 v = *(const float4*)(h2 + (long long)r * 16 + q * 4);  // rows are 16-wide, 16B aligned
    float* dst = out + (long long)c * 7 + q * 4;
    atomicAdd(dst + 0, w * v.x);
    atomicAdd(dst + 1, w * v.y);
    atomicAdd(dst + 2, w * v.z);
    if (q == 0) atomicAdd(dst + 3, w * v.w);   // feature 7..15 don't exist; q=1 covers f=4..6
}

extern "C" void kernel_launch(void* const* d_in, const int* in_sizes, int n_in,
                              void* d_out, int out_size, void* d_ws, size_t ws_size,
                              hipStream_t stream) {
    const float* x  = (const float*)d_in[0];
    const int*   ei = (const int*)d_in[1];
    const float* W1 = (const float*)d_in[2];
    const float* b1 = (const float*)d_in[3];
    const float* W2 = (const float*)d_in[4];
    const float* b2 = (const float*)d_in[5];
    float* out = (float*)d_out;

    const int* row = ei;          // edge_index[0]
    const int* col = ei + NE;     // edge_index[1]

    char* ws = (char*)d_ws;
    float* dis  = (float*)(ws);                                   //  1.0 MB: deg -> dis
    float* bufA = (float*)(ws + (size_t)(1u << 20));              // 16.0 MB: h1, then hrelu
    float* bufB = (float*)(ws + (size_t)(1u << 20) + (16u << 20));// 16.0 MB: agg1, then h2

    const int T = 256;
    dim3 blk(T);

    // degrees + normalization
    k_deg_init <<<(NN + T - 1) / T, blk, 0, stream>>>(dis);
    k_deg_count<<<(NE + T - 1) / T, blk, 0, stream>>>(col, dis);
    k_dis      <<<(NN + T - 1) / T, blk, 0, stream>>>(dis);

    // layer 1
    int wmma_blocks = (NTILE * 32 + 127) / 128;   // 4 waves (tiles) per 128-thread block
    k_lin1_wmma<<<wmma_blocks, dim3(128), 0, stream>>>(x, W1, bufA);
    k_agg1_init<<<(NN * 16 + T - 1) / T, blk, 0, stream>>>(bufA, dis, bufB);
    k_edge1    <<<(int)(((long long)NE * 4 + T - 1) / T), blk, 0, stream>>>(row, col, dis, bufA, bufB);
    k_relu1    <<<(NN * 16 + T - 1) / T, blk, 0, stream>>>(bufB, b1, bufA);   // bufA now hrelu

    // layer 2
    k_lin2_wmma<<<wmma_blocks, dim3(128), 0, stream>>>(bufA, W2, bufB);       // bufB now h2 (16-wide, padded)
    k_out_init <<<(NN * 7 + T - 1) / T, blk, 0, stream>>>(bufB, dis, b2, out);
    k_edge2    <<<(int)(((long long)NE * 2 + T - 1) / T), blk, 0, stream>>>(row, col, dis, bufB, out);
}